// DGCN_73770358276514
// MI455X (gfx1250) — compile-verified
//
#include <hip/hip_runtime.h>

// DGCN forward for MI455X (gfx1250), wave32, f32 WMMA 16x16x4 + TDM staging.
// B=16, C=64, N=1024, L=12.
#define B_   16
#define C_   64
#define N_   1024
#define L_   12
#define NL_  (N_ * L_)      // 12288
#define KTOP 819            // int(1024*0.8)
#define SCALE 0.125f        // 1/sqrt(C)

typedef float v2f __attribute__((ext_vector_type(2)));
typedef float v8f __attribute__((ext_vector_type(8)));

// D = A(16x4,f32) * B(4x16,f32) + C(16x16,f32)   -> v_wmma_f32_16x16x4_f32
__device__ __forceinline__ v8f wmma4(v2f a, v2f b, v8f c) {
  return __builtin_amdgcn_wmma_f32_16x16x4_f32(false, a, false, b, (short)0, c,
                                               false, false);
}

// ---------------------------------------------------------------------------
// Tensor Data Mover: 2D f32 tile (tile_w x tile_h) from global (row stride
// gstride elements) into LDS with +1 dword pad per 64 dwords (row stride 65).
// Descriptor layout per CDNA5 ISA. All args must be
// block-uniform (SGPRs). Issue from one wave only; publish via barrier.
// This toolchain's builtin takes the 6-arg (clang-23 / therock) form:
//   (uint32x4 g0, int32x8 g1, int32x4, int32x4, int32x8, i32 cpol)
// ---------------------------------------------------------------------------
#if __has_builtin(__builtin_amdgcn_tensor_load_to_lds) && \
    __has_builtin(__builtin_amdgcn_s_wait_tensorcnt)
#define HAVE_TDM 1
typedef unsigned int v4u __attribute__((ext_vector_type(4)));
typedef int v4i __attribute__((ext_vector_type(4)));
typedef int v8i __attribute__((ext_vector_type(8)));

__device__ __forceinline__ void tdm_load_tile_f32(unsigned lds_off,
                                                  const float* gaddr,
                                                  unsigned tensor_w,
                                                  unsigned tensor_h,
                                                  unsigned gstride,
                                                  unsigned tile_w,
                                                  unsigned tile_h) {
  unsigned long long ga = (unsigned long long)(size_t)gaddr;
  v4u g0;
  g0[0] = 1u;                                   // count=1, user mode, no gather
  g0[1] = lds_off;                              // lds_addr (bytes)
  g0[2] = (unsigned)ga;                         // global_addr[31:0]
  g0[3] = (unsigned)((ga >> 32) & 0x1FFFFFFu)   // global_addr[56:32]
          | (2u << 30);                         // type = 2 ("image")
  v8i g1;
  // data_size=4B (2<<16) | pad_enable (1<<20) | pad_interval=64dw (5<<22)
  // | pad_amount=1dw (0<<25); workgroup_mask=0 (not in a cluster)
  g1[0] = (2 << 16) | (1 << 20) | (5 << 22);
  g1[1] = (int)((tensor_w & 0xFFFFu) << 16);                    // dim0[15:0]
  g1[2] = (int)((tensor_w >> 16) | ((tensor_h & 0xFFFFu) << 16));
  g1[3] = (int)((tensor_h >> 16) | (tile_w << 16));             // tile_dim0
  g1[4] = (int)(tile_h & 0xFFFFu);              // tile_dim1 (tile_dim2 = 0)
  g1[5] = (int)gstride;                         // tensor_dim0_stride[31:0]
  g1[6] = 0;                                    // stride hi / dim1_stride lo
  g1[7] = 0;
  v4i z4 = {0, 0, 0, 0};                        // 2D: groups 2/3 unused
  v8i z8 = {0, 0, 0, 0, 0, 0, 0, 0};
  __builtin_amdgcn_tensor_load_to_lds(g0, g1, z4, z4, z8, 0);
}
#else
#define HAVE_TDM 0
#endif

// ---------------------------------------------------------------------------
// Kernel 1: x1[b,o,j] = sum_c W[o,c]*x[b,c,j] + bias[o],  j = n*L+l (NL=12288)
// grid (NL/64, B), block 256 (8 waves). Per block: 64x64 out tile, K=64.
// Dynamic LDS: sW [64x65] @0, sX [64x65] @16640B (TDM needs the byte offset).
// ---------------------------------------------------------------------------
__global__ void __launch_bounds__(256)
k_conv1(const float* __restrict__ x, const float* __restrict__ W,
        const float* __restrict__ bias, float* __restrict__ x1) {
  extern __shared__ float sm[];
  float (*sW)[65] = (float(*)[65])sm;
  float (*sX)[65] = (float(*)[65])(sm + 64 * 65);
  const int tid = threadIdx.x;
  const int b  = blockIdx.y;
  const int j0 = blockIdx.x * 64;
  const int wv = tid >> 5, lane = tid & 31;

#if HAVE_TDM
  if (wv == 0)
    tdm_load_tile_f32((unsigned)((char*)sX - (char*)sm),
                      x + (size_t)(b * C_) * NL_ + j0, NL_, C_, NL_, 64, 64);
#endif
  for (int e = tid; e < 4096; e += 256) sW[e >> 6][e & 63] = W[e];
#if HAVE_TDM
  if (wv == 0) __builtin_amdgcn_s_wait_tensorcnt(0);
#else
  for (int e = tid; e < 4096; e += 256) {
    int c = e >> 6, j = e & 63;
    sX[c][j] = x[(b * C_ + c) * NL_ + j0 + j];
  }
#endif
  __syncthreads();

  const int hi = lane >> 4, lm = lane & 15;
  const int rt = wv >> 1;            // row tile 0..3 (channels)
  const int ct0 = (wv & 1) * 2;      // this wave owns col tiles ct0, ct0+1

  v8f acc[2];
  for (int t = 0; t < 2; ++t)
    for (int v = 0; v < 8; ++v) acc[t][v] = bias[rt * 16 + v + hi * 8];

  for (int kk = 0; kk < 16; ++kk) {
    v2f a;
    a.x = sW[rt * 16 + lm][kk * 4 + hi * 2];
    a.y = sW[rt * 16 + lm][kk * 4 + hi * 2 + 1];
    for (int t = 0; t < 2; ++t) {
      int ct = ct0 + t;
      v2f bf;
      bf.x = sX[kk * 4 + hi * 2][ct * 16 + lm];
      bf.y = sX[kk * 4 + hi * 2 + 1][ct * 16 + lm];
      acc[t] = wmma4(a, bf, acc[t]);
    }
  }
  for (int t = 0; t < 2; ++t) {
    int ct = ct0 + t;
    for (int v = 0; v < 8; ++v) {
      int o = rt * 16 + v + hi * 8;
      x1[(b * C_ + o) * NL_ + j0 + ct * 16 + lm] = acc[t][v];
    }
  }
}

// ---------------------------------------------------------------------------
// Kernel 2: xs[b,c,n] = sum_l x1[b,c,n,l]   (3x float4 loads per thread)
// ---------------------------------------------------------------------------
__global__ void __launch_bounds__(256)
k_sumL(const float* __restrict__ x1, float* __restrict__ xs) {
  int idx = blockIdx.x * 256 + threadIdx.x;   // over B*C*N
  const float4* p = (const float4*)(x1 + (size_t)idx * L_);  // 48B stride
  float4 q0 = p[0], q1 = p[1], q2 = p[2];
  xs[idx] = q0.x + q0.y + q0.z + q0.w + q1.x + q1.y + q1.z + q1.w +
            q2.x + q2.y + q2.z + q2.w;
}

// ---------------------------------------------------------------------------
// Kernel 3: fused adjacency. grid (N/16, B), block 256.
// S1 = relu(scale * XsRowsᵀ @ memory), S2 = relu(scale * XsRowsᵀ @ Xs)
// -> rowwise softmax each -> adj = softmax(w00*a1 + w01*a2 + b0)
// -> top-819 mask via per-row threshold bisection -> store adj.
// ---------------------------------------------------------------------------
__global__ void __launch_bounds__(256)
k_adj(const float* __restrict__ xs, const float* __restrict__ mem,
      const float* __restrict__ fcw, const float* __restrict__ fcb,
      float* __restrict__ adj) {
  extern __shared__ float sm[];
  float (*s1)[1025] = (float(*)[1025])sm;
  float (*s2)[1025] = (float(*)[1025])(sm + 16 * 1025);
  float (*sA)[65]   = (float(*)[65])(sm + 2 * 16 * 1025);
  float (*sBm)[65]  = (float(*)[65])(sm + 2 * 16 * 1025 + 16 * 65);
  float (*sBx)[65]  = (float(*)[65])(sm + 2 * 16 * 1025 + 16 * 65 + 64 * 65);

  const int tid = threadIdx.x;
  const int b  = blockIdx.y;
  const int n0 = blockIdx.x * 16;

  for (int e = tid; e < 1024; e += 256) {    // A[r][c] = xs[b,c,n0+r]
    int c = e >> 4, r = e & 15;
    sA[r][c] = xs[(b * C_ + c) * N_ + n0 + r];
  }

  const int wv = tid >> 5, lane = tid & 31, hi = lane >> 4, lm = lane & 15;
  const int mat = wv >> 2, ct = wv & 3;      // 4 waves -> S1 tiles, 4 -> S2

  for (int ch = 0; ch < 16; ++ch) {          // 64-wide column chunks
    int m0 = ch * 64;
#if HAVE_TDM
    if (wv == 0) {
      tdm_load_tile_f32((unsigned)((char*)sBm - (char*)sm), mem + m0,
                        N_, C_, N_, 64, 64);
      tdm_load_tile_f32((unsigned)((char*)sBx - (char*)sm),
                        xs + (size_t)(b * C_) * N_ + m0, N_, C_, N_, 64, 64);
      __builtin_amdgcn_s_wait_tensorcnt(0);
    }
#else
    for (int e = tid; e < 4096; e += 256) {
      int c = e >> 6, m = e & 63;
      sBm[c][m] = mem[c * N_ + m0 + m];
      sBx[c][m] = xs[(b * C_ + c) * N_ + m0 + m];
    }
#endif
    __syncthreads();

    v8f acc = {};
    for (int kk = 0; kk < 16; ++kk) {        // K = C = 64
      v2f a;
      a.x = sA[lm][kk * 4 + hi * 2];
      a.y = sA[lm][kk * 4 + hi * 2 + 1];
      v2f bf;
      if (mat == 0) {                        // uniform per wave: EXEC full
        bf.x = sBm[kk * 4 + hi * 2][ct * 16 + lm];
        bf.y = sBm[kk * 4 + hi * 2 + 1][ct * 16 + lm];
      } else {
        bf.x = sBx[kk * 4 + hi * 2][ct * 16 + lm];
        bf.y = sBx[kk * 4 + hi * 2 + 1][ct * 16 + lm];
      }
      acc = wmma4(a, bf, acc);
    }
    float (*dst)[1025] = mat ? s2 : s1;
    for (int v = 0; v < 8; ++v) {
      float t = acc[v] * SCALE;
      dst[v + hi * 8][m0 + ct * 16 + lm] = t > 0.f ? t : 0.f;   // relu
    }
    __syncthreads();
  }

  // 16 threads per row; shfl_xor masks < 16 stay within each 16-lane half.
  const int row = tid >> 4;
  const int sub = tid & 15;

  for (int m = 0; m < 2; ++m) {              // softmax rows of S1 and S2
    float (*s)[1025] = m ? s2 : s1;
    float mx = -1e30f;
    for (int i = 0; i < 64; ++i) mx = fmaxf(mx, s[row][sub + 16 * i]);
    for (int d = 8; d >= 1; d >>= 1) mx = fmaxf(mx, __shfl_xor(mx, d, 32));
    float sum = 0.f;
    for (int i = 0; i < 64; ++i) {
      float e = __expf(s[row][sub + 16 * i] - mx);
      s[row][sub + 16 * i] = e;
      sum += e;
    }
    for (int d = 8; d >= 1; d >>= 1) sum += __shfl_xor(sum, d, 32);
    float inv = 1.f / sum;
    for (int i = 0; i < 64; ++i) s[row][sub + 16 * i] *= inv;
  }

  const float w00 = fcw[0], w01 = fcw[1], b0f = fcb[0];
  {  // combine + second softmax, result into s1
    float mx = -1e30f;
    for (int i = 0; i < 64; ++i) {
      float t = w00 * s1[row][sub + 16 * i] + w01 * s2[row][sub + 16 * i] + b0f;
      s1[row][sub + 16 * i] = t;
      mx = fmaxf(mx, t);
    }
    for (int d = 8; d >= 1; d >>= 1) mx = fmaxf(mx, __shfl_xor(mx, d, 32));
    float sum = 0.f;
    for (int i = 0; i < 64; ++i) {
      float e = __expf(s1[row][sub + 16 * i] - mx);
      s1[row][sub + 16 * i] = e;
      sum += e;
    }
    for (int d = 8; d >= 1; d >>= 1) sum += __shfl_xor(sum, d, 32);
    float inv = 1.f / sum;
    for (int i = 0; i < 64; ++i) s1[row][sub + 16 * i] *= inv;
  }

  // top-819 mask: bisect threshold T in (0,1] s.t. count(v >= T) == KTOP
  float lo = 0.f, hiv = 1.0f;
  for (int it = 0; it < 28; ++it) {
    float mid = 0.5f * (lo + hiv);
    int cnt = 0;
    for (int i = 0; i < 64; ++i) cnt += (s1[row][sub + 16 * i] >= mid) ? 1 : 0;
    for (int d = 8; d >= 1; d >>= 1) cnt += __shfl_xor(cnt, d, 32);
    if (cnt > KTOP) lo = mid; else hiv = mid;
  }
  for (int i = 0; i < 64; ++i)
    if (s1[row][sub + 16 * i] < hiv) s1[row][sub + 16 * i] = 0.f;
  __syncthreads();

  for (int e = tid; e < 16 * 1024; e += 256) {   // coalesced store
    int r = e >> 10, j = e & 1023;
    adj[(b * N_ + n0 + r) * N_ + j] = s1[r][j];
  }
}

// ---------------------------------------------------------------------------
// Kernel 4 (dominant, fused): for fixed (b,l), 64-col tile m0:
//   x2[c,m] = sum_n x1[b,c,n,l] * adj[b,n,m]        (64 x 1024 x 64 GEMM)
//   x3      = W @ x2 + bias                          (64 x 64 x 64 GEMM)
//   out     = x3 * emb[m] + x[b,c,m,l]
// grid (N/64, L, B), block 256. x2 never touches HBM. adj tiles stream in via
// TDM, double-buffered: issue chunk nc+1, s_wait_tensorcnt(1), compute nc.
// ---------------------------------------------------------------------------
__global__ void __launch_bounds__(256)
k_diffuse(const float* __restrict__ x1, const float* __restrict__ adj,
          const float* __restrict__ W, const float* __restrict__ bias,
          const float* __restrict__ emb, const float* __restrict__ x0,
          float* __restrict__ out) {
  extern __shared__ float sm[];
  float (*sW)[65]  = (float(*)[65])sm;
  float (*sA)[65]  = (float(*)[65])(sm + 64 * 65);
  float (*sB0)[65] = (float(*)[65])(sm + 2 * 64 * 65);
  float (*sB1)[65] = (float(*)[65])(sm + 3 * 64 * 65);
  float (*sX2)[65] = (float(*)[65])(sm + 4 * 64 * 65);

  const int tid = threadIdx.x;
  const int m0 = blockIdx.x * 64;
  const int l  = blockIdx.y;
  const int b  = blockIdx.z;
  const int wv = tid >> 5, lane = tid & 31, hi = lane >> 4, lm = lane & 15;
  const int rt = wv >> 1, ct0 = (wv & 1) * 2;

#if HAVE_TDM
  if (wv == 0)   // prime the pipeline: chunk 0 -> sB0
    tdm_load_tile_f32((unsigned)((char*)sB0 - (char*)sm),
                      adj + (size_t)(b * N_) * N_ + m0, N_, N_, N_, 64, 64);
#endif
  for (int e = tid; e < 4096; e += 256) sW[e >> 6][e & 63] = W[e];

  v8f acc[2] = {{}, {}};
  for (int nc = 0; nc < 16; ++nc) {          // K = N = 1024 in 64-chunks
    int nb = nc * 64;
    float (*sB)[65] = (nc & 1) ? sB1 : sB0;
    for (int e = tid; e < 4096; e += 256) {
      int c = e >> 6, nn = e & 63;
      sA[c][nn] = x1[(b * C_ + c) * NL_ + (nb + nn) * L_ + l];
    }
#if HAVE_TDM
    if (wv == 0) {
      if (nc + 1 < 16) {                     // prefetch next adj chunk via TDM
        float (*sBn)[65] = (nc & 1) ? sB0 : sB1;
        tdm_load_tile_f32((unsigned)((char*)sBn - (char*)sm),
                          adj + (size_t)(b * N_ + nb + 64) * N_ + m0,
                          N_, N_, N_, 64, 64);
        __builtin_amdgcn_s_wait_tensorcnt(1);  // current done, next in flight
      } else {
        __builtin_amdgcn_s_wait_tensorcnt(0);
      }
    }
    if (nc + 1 < 16)                         // cache hint for next x1 chunk
      __builtin_prefetch(
          &x1[(b * C_ + (tid >> 2)) * NL_ + (nb + 64 + (tid & 3) * 16) * L_ + l],
          0, 1);
#else
    for (int e = tid; e < 4096; e += 256) {
      int nn = e >> 6, m = e & 63;
      sB[nn][m] = adj[(b * N_ + nb + nn) * N_ + m0 + m];
    }
    if (nc + 1 < 16)
      __builtin_prefetch(
          &adj[(b * N_ + nb + 64 + (tid >> 2)) * N_ + m0 + (tid & 3) * 16], 0, 1);
#endif
    __syncthreads();
    for (int kk = 0; kk < 16; ++kk) {
      v2f a;
      a.x = sA[rt * 16 + lm][kk * 4 + hi * 2];
      a.y = sA[rt * 16 + lm][kk * 4 + hi * 2 + 1];
      for (int t = 0; t < 2; ++t) {
        int ct = ct0 + t;
        v2f bf;
        bf.x = sB[kk * 4 + hi * 2][ct * 16 + lm];
        bf.y = sB[kk * 4 + hi * 2 + 1][ct * 16 + lm];
        acc[t] = wmma4(a, bf, acc[t]);
      }
    }
    __syncthreads();                         // before sA/sB reuse next iter
  }

  // stage x2 tile to LDS for the channel-mixing conv
  for (int t = 0; t < 2; ++t) {
    int ct = ct0 + t;
    for (int v = 0; v < 8; ++v)
      sX2[rt * 16 + v + hi * 8][ct * 16 + lm] = acc[t][v];
  }
  __syncthreads();

  v8f acc2[2];
  for (int t = 0; t < 2; ++t)
    for (int v = 0; v < 8; ++v) acc2[t][v] = bias[rt * 16 + v + hi * 8];
  for (int kk = 0; kk < 16; ++kk) {
    v2f a;
    a.x = sW[rt * 16 + lm][kk * 4 + hi * 2];
    a.y = sW[rt * 16 + lm][kk * 4 + hi * 2 + 1];
    for (int t = 0; t < 2; ++t) {
      int ct = ct0 + t;
      v2f bf;
      bf.x = sX2[kk * 4 + hi * 2][ct * 16 + lm];
      bf.y = sX2[kk * 4 + hi * 2 + 1][ct * 16 + lm];
      acc2[t] = wmma4(a, bf, acc2[t]);
    }
  }

  for (int t = 0; t < 2; ++t) {
    int ct = ct0 + t;
    int m = m0 + ct * 16 + lm;
    float em = emb[m];
    for (int v = 0; v < 8; ++v) {
      int c = rt * 16 + v + hi * 8;
      size_t idx = ((size_t)(b * C_ + c) * N_ + m) * L_ + l;
      out[idx] = acc2[t][v] * em + x0[idx];
    }
  }
}

// ---------------------------------------------------------------------------
extern "C" void kernel_launch(void* const* d_in, const int* in_sizes, int n_in,
                              void* d_out, int out_size, void* d_ws, size_t ws_size,
                              hipStream_t stream) {
  (void)in_sizes; (void)n_in; (void)out_size; (void)ws_size;
  const float* x    = (const float*)d_in[0];
  const float* W    = (const float*)d_in[1];
  const float* bias = (const float*)d_in[2];
  const float* mem  = (const float*)d_in[3];
  const float* fcw  = (const float*)d_in[4];
  const float* fcb  = (const float*)d_in[5];
  const float* emb  = (const float*)d_in[6];
  float* out = (float*)d_out;

  // workspace layout (floats): x1 [B*C*NL], xs [B*C*N], adj [B*N*N]  (~118 MB)
  float* x1w  = (float*)d_ws;
  float* xsw  = x1w + (size_t)B_ * C_ * NL_;
  float* adjw = xsw + (size_t)B_ * C_ * N_;

  size_t sm1 = (size_t)(2 * 64 * 65) * sizeof(float);
  k_conv1<<<dim3(NL_ / 64, B_), 256, sm1, stream>>>(x, W, bias, x1w);
  k_sumL<<<dim3((B_ * C_ * N_) / 256), 256, 0, stream>>>(x1w, xsw);

  size_t sm3 = (size_t)(2 * 16 * 1025 + 16 * 65 + 2 * 64 * 65) * sizeof(float);
  k_adj<<<dim3(N_ / 16, B_), 256, sm3, stream>>>(xsw, mem, fcw, fcb, adjw);

  size_t sm4 = (size_t)(5 * 64 * 65) * sizeof(float);
  k_diffuse<<<dim3(N_ / 64, L_, B_), 256, sm4, stream>>>(x1w, adjw, W, bias,
                                                         emb, x, out);
}